// StateSpaceModel_32452772889244
// MI455X (gfx1250) — compile-verified
//
#include <hip/hip_runtime.h>
#include <hip/hip_bf16.h>
#include <math.h>

typedef float v2f __attribute__((ext_vector_type(2)));
typedef float v8f __attribute__((ext_vector_type(8)));

// Problem constants (fixed by reference)
#define BATCH   2
#define SEQ     2048
#define DMODEL  1024
#define DTRANK  64
#define NSTATE  16
#define KPROJ   (DTRANK + 2 * NSTATE)   // 96
#define MROWS   (BATCH * SEQ)           // 4096
#define NCHAN   (BATCH * DMODEL)        // 2048
#define NCHUNK  32
#define CLEN    (SEQ / NCHUNK)          // 64

// ---------------------------------------------------------------------------
// GEMM1: xp[m][j] = sum_k u[m][k] * W1[j][k]   (M=4096, N=96, K=1024)
// one 16x16 tile per wave, V_WMMA_F32_16X16X4_F32 over K
// ---------------------------------------------------------------------------
__global__ __launch_bounds__(256)
void gemm_xproj(const float* __restrict__ U, const float* __restrict__ W1,
                float* __restrict__ xp) {
    const int lane  = threadIdx.x & 31;
    const int wave  = threadIdx.x >> 5;
    const int tile  = blockIdx.x * (blockDim.x >> 5) + wave;     // 0..1535
    const int tilesN = KPROJ / 16;                               // 6
    const int m0 = (tile / tilesN) * 16;
    const int n0 = (tile % tilesN) * 16;

    const int row   = m0 + (lane & 15);
    const int col   = n0 + (lane & 15);
    const int khalf = (lane >> 4) * 2;     // lanes 0-15 -> K+{0,1}, 16-31 -> K+{2,3}

    v8f acc = {};
    #pragma unroll 4
    for (int k = 0; k < DMODEL; k += 4) {
        v2f a = *(const v2f*)(U  + row * DMODEL + k + khalf);
        v2f b = *(const v2f*)(W1 + col * DMODEL + k + khalf);
        acc = __builtin_amdgcn_wmma_f32_16x16x4_f32(false, a, false, b,
                                                    (short)0, acc, false, false);
    }
    const int rbase = m0 + ((lane >> 4) ? 8 : 0);
    #pragma unroll
    for (int i = 0; i < 8; ++i)
        xp[(rbase + i) * KPROJ + col] = acc[i];
}

// ---------------------------------------------------------------------------
// GEMM2: delta[m][n] = softplus(sum_k xp[m][k]*W2[n][k] + bias[n])
//        (M=4096, N=1024, K=64; A ld = 96, B ld = 64)
// ---------------------------------------------------------------------------
__global__ __launch_bounds__(256)
void gemm_delta(const float* __restrict__ xp, const float* __restrict__ W2,
                const float* __restrict__ bias, float* __restrict__ delta) {
    const int lane  = threadIdx.x & 31;
    const int wave  = threadIdx.x >> 5;
    const int tile  = blockIdx.x * (blockDim.x >> 5) + wave;     // 0..16383
    const int tilesN = DMODEL / 16;                              // 64
    const int m0 = (tile / tilesN) * 16;
    const int n0 = (tile % tilesN) * 16;

    const int row   = m0 + (lane & 15);
    const int col   = n0 + (lane & 15);
    const int khalf = (lane >> 4) * 2;

    v8f acc = {};
    #pragma unroll
    for (int k = 0; k < DTRANK; k += 4) {
        v2f a = *(const v2f*)(xp + row * KPROJ  + k + khalf);
        v2f b = *(const v2f*)(W2 + col * DTRANK + k + khalf);
        acc = __builtin_amdgcn_wmma_f32_16x16x4_f32(false, a, false, b,
                                                    (short)0, acc, false, false);
    }
    const int   rbase = m0 + ((lane >> 4) ? 8 : 0);
    const float bn    = bias[col];
    #pragma unroll
    for (int i = 0; i < 8; ++i) {
        float x  = acc[i] + bn;
        float sp = (x > 20.0f) ? x : log1pf(__expf(x));   // softplus
        delta[(rbase + i) * DMODEL + col] = sp;
    }
}

// ---------------------------------------------------------------------------
// Pass 1: per-chunk local scan (zero entry state).
// Writes local y to out, and chunk aggregates (prodA = exp(A*sum(delta)),
// final local state) to agg in SoA layout agg[(chunk*32 + j)*NCHAN + ch].
// One lane per channel; 16 states in VGPRs.
// ---------------------------------------------------------------------------
__global__ __launch_bounds__(256)
void scan_local(const float* __restrict__ U, const float* __restrict__ A_log,
                const float* __restrict__ xp, const float* __restrict__ delta,
                float* __restrict__ out, float* __restrict__ agg) {
    const int ch = blockIdx.x * blockDim.x + threadIdx.x;   // 0..2047
    const int c  = blockIdx.y;                              // chunk
    const int bb = ch >> 10;
    const int dd = ch & (DMODEL - 1);

    float Av[NSTATE];
    #pragma unroll
    for (int n = 0; n < NSTATE; ++n) Av[n] = -__expf(A_log[n]);

    float st[NSTATE];
    #pragma unroll
    for (int n = 0; n < NSTATE; ++n) st[n] = 0.0f;

    float dsum = 0.0f;
    const int t0 = c * CLEN;
    for (int t = t0; t < t0 + CLEN; ++t) {
        const int rowu = bb * SEQ + t;
        const float dl = delta[rowu * DMODEL + dd];
        const float uv = U[rowu * DMODEL + dd];
        // B = xp[row, 64:80], C = xp[row, 80:96]  (uniform address -> broadcast)
        const float4* Bp = (const float4*)(xp + rowu * KPROJ + DTRANK);
        const float4* Cp = (const float4*)(xp + rowu * KPROJ + DTRANK + NSTATE);
        float Bv[NSTATE], Cv[NSTATE];
        #pragma unroll
        for (int q = 0; q < 4; ++q) {
            float4 vb = Bp[q], vc = Cp[q];
            Bv[4*q+0]=vb.x; Bv[4*q+1]=vb.y; Bv[4*q+2]=vb.z; Bv[4*q+3]=vb.w;
            Cv[4*q+0]=vc.x; Cv[4*q+1]=vc.y; Cv[4*q+2]=vc.z; Cv[4*q+3]=vc.w;
        }
        const float du = dl * uv;
        float y = 0.0f;
        #pragma unroll
        for (int n = 0; n < NSTATE; ++n) {
            float dA = __expf(dl * Av[n]);
            st[n] = fmaf(dA, st[n], du * Bv[n]);
            y     = fmaf(st[n], Cv[n], y);
        }
        dsum += dl;
        out[rowu * DMODEL + dd] = y;    // local contribution; fixed up in pass 2
    }
    // aggregates: prod(dA) over chunk = exp(Av[n] * sum(delta))
    const int base = (c * 2 * NSTATE) * NCHAN + ch;
    #pragma unroll
    for (int n = 0; n < NSTATE; ++n)
        agg[base + n * NCHAN] = __expf(Av[n] * dsum);
    #pragma unroll
    for (int n = 0; n < NSTATE; ++n)
        agg[base + (NSTATE + n) * NCHAN] = st[n];
}

// ---------------------------------------------------------------------------
// Pass 2: combine chunk prefixes, replay chunk adding correction
//   y_t += C_t . (cumprod(dA)_t * entry)   and the skip term u*D.
// ---------------------------------------------------------------------------
__global__ __launch_bounds__(256)
void scan_fixup(const float* __restrict__ U, const float* __restrict__ A_log,
                const float* __restrict__ xp, const float* __restrict__ delta,
                const float* __restrict__ Dvec, const float* __restrict__ agg,
                float* __restrict__ out) {
    const int ch = blockIdx.x * blockDim.x + threadIdx.x;
    const int c  = blockIdx.y;
    const int bb = ch >> 10;
    const int dd = ch & (DMODEL - 1);

    float Av[NSTATE];
    #pragma unroll
    for (int n = 0; n < NSTATE; ++n) Av[n] = -__expf(A_log[n]);

    // entry state = scan over aggregates of chunks < c (coalesced SoA loads)
    float entry[NSTATE];
    #pragma unroll
    for (int n = 0; n < NSTATE; ++n) entry[n] = 0.0f;
    for (int cc = 0; cc < c; ++cc) {
        const int base = (cc * 2 * NSTATE) * NCHAN + ch;
        #pragma unroll
        for (int n = 0; n < NSTATE; ++n) {
            float p = agg[base + n * NCHAN];
            float f = agg[base + (NSTATE + n) * NCHAN];
            entry[n] = fmaf(p, entry[n], f);
        }
    }

    const float Dd = Dvec[dd];
    const int t0 = c * CLEN;
    for (int t = t0; t < t0 + CLEN; ++t) {
        const int rowu = bb * SEQ + t;
        const float dl = delta[rowu * DMODEL + dd];
        const float uv = U[rowu * DMODEL + dd];
        const float4* Cp = (const float4*)(xp + rowu * KPROJ + DTRANK + NSTATE);
        float Cv[NSTATE];
        #pragma unroll
        for (int q = 0; q < 4; ++q) {
            float4 vc = Cp[q];
            Cv[4*q+0]=vc.x; Cv[4*q+1]=vc.y; Cv[4*q+2]=vc.z; Cv[4*q+3]=vc.w;
        }
        float ycorr = 0.0f;
        #pragma unroll
        for (int n = 0; n < NSTATE; ++n) {
            entry[n] *= __expf(dl * Av[n]);       // running cumprod(dA)*entry
            ycorr = fmaf(entry[n], Cv[n], ycorr);
        }
        const int idx = rowu * DMODEL + dd;
        out[idx] = out[idx] + ycorr + uv * Dd;
    }
}

// ---------------------------------------------------------------------------
extern "C" void kernel_launch(void* const* d_in, const int* in_sizes, int n_in,
                              void* d_out, int out_size, void* d_ws, size_t ws_size,
                              hipStream_t stream) {
    const float* U     = (const float*)d_in[0];   // (2,2048,1024)
    const float* A_log = (const float*)d_in[1];   // (16,)
    const float* W1    = (const float*)d_in[2];   // (96,1024)
    const float* W2    = (const float*)d_in[3];   // (1024,64)
    const float* bias  = (const float*)d_in[4];   // (1024,)
    const float* Dvec  = (const float*)d_in[5];   // (1024,)
    float* out = (float*)d_out;

    float* ws    = (float*)d_ws;
    float* xp    = ws;                                    // 4096*96
    float* delta = xp + (size_t)MROWS * KPROJ;            // 4096*1024
    float* agg   = delta + (size_t)MROWS * DMODEL;        // 32*32*2048

    // GEMM1: 4096/16 * 96/16 = 1536 tiles, 8 waves/block
    gemm_xproj<<<dim3(1536 / 8), dim3(256), 0, stream>>>(U, W1, xp);

    // GEMM2: 4096/16 * 1024/16 = 16384 tiles
    gemm_delta<<<dim3(16384 / 8), dim3(256), 0, stream>>>(xp, W2, bias, delta);

    // Chunked parallel scan: 2048 channels x 32 chunks
    dim3 sgrid(NCHAN / 256, NCHUNK);
    scan_local<<<sgrid, dim3(256), 0, stream>>>(U, A_log, xp, delta, out, agg);
    scan_fixup<<<sgrid, dim3(256), 0, stream>>>(U, A_log, xp, delta, Dvec, agg, out);
}